// SingleBatchRasterization_85091892068505
// MI455X (gfx1250) — compile-verified
//
#include <hip/hip_runtime.h>
#include <hip/hip_bf16.h>
#include <math.h>

#define NGAUSS 2048
#define NVIEW  2
#define IMGH   128
#define IMGW   128
#define EPS2D  0.3f
#define A_MIN  (1.0f/255.0f)
#define A_MAX  0.999f
#define ZNEARF 0.01f
#define ZFARF  100.0f
#define NLOG2E (-1.4426950408889634f)   // -log2(e): fold exp->exp2 into coefficients

typedef float v2f __attribute__((ext_vector_type(2)));
typedef float v8f __attribute__((ext_vector_type(8)));
typedef unsigned int v4u __attribute__((ext_vector_type(4)));
typedef int v4i_t __attribute__((ext_vector_type(4)));
typedef int v8i_t __attribute__((ext_vector_type(8)));

// workspace layout (in floats)
#define U_NPARAM 11   // mx,my,ca,cb,cc,opac,valid,r,g,b,z
#define S_NPARAM 10   // v0..v5 (pre-scaled by -log2e), r,g,b,z
#define OFF_U    0
#define OFF_IDX  (NVIEW*U_NPARAM*NGAUSS)           // int region
#define OFF_S    (OFF_IDX + NVIEW*NGAUSS)

// ---------------------------------------------------------------------------
// Kernel 1: per (view, gaussian) projection -> unsorted SoA params
// ---------------------------------------------------------------------------
__global__ void prep_kernel(const float* __restrict__ xyz, const float* __restrict__ rgb,
                            const float* __restrict__ scale, const float* __restrict__ rot,
                            const float* __restrict__ opacity, const float* __restrict__ tin,
                            const float* __restrict__ cov_t, const float* __restrict__ ms3,
                            const float* __restrict__ w2cs, const float* __restrict__ Ks,
                            const float* __restrict__ ts, float* __restrict__ U)
{
    int gid = blockIdx.x * blockDim.x + threadIdx.x;
    if (gid >= NVIEW * NGAUSS) return;
    int v = gid / NGAUSS, n = gid % NGAUSS;

    float dt = ts[v] - tin[n];
    float m0 = xyz[n*3+0] + ms3[n*3+0] * dt;
    float m1 = xyz[n*3+1] + ms3[n*3+1] * dt;
    float m2 = xyz[n*3+2] + ms3[n*3+2] * dt;
    float marg = __expf(-0.5f * dt * dt / cov_t[n]);
    float op = opacity[n] * marg;

    const float* vm = w2cs + v*16;
    float Rm[3][3] = {{vm[0],vm[1],vm[2]},{vm[4],vm[5],vm[6]},{vm[8],vm[9],vm[10]}};
    float tc[3] = {vm[3], vm[7], vm[11]};
    float xc = Rm[0][0]*m0 + Rm[0][1]*m1 + Rm[0][2]*m2 + tc[0];
    float yc = Rm[1][0]*m0 + Rm[1][1]*m1 + Rm[1][2]*m2 + tc[1];
    float zc = Rm[2][0]*m0 + Rm[2][1]*m1 + Rm[2][2]*m2 + tc[2];

    const float* Km = Ks + v*9;
    float fx = Km[0], fy = Km[4], cx = Km[2], cy = Km[5];

    // quaternion (w,x,y,z) -> rotation
    float qw = rot[n*4+0], qx = rot[n*4+1], qy = rot[n*4+2], qz = rot[n*4+3];
    float qn = rsqrtf(qw*qw + qx*qx + qy*qy + qz*qz);
    qw *= qn; qx *= qn; qy *= qn; qz *= qn;
    float G[3][3];
    G[0][0] = 1.f - 2.f*(qy*qy + qz*qz); G[0][1] = 2.f*(qx*qy - qw*qz); G[0][2] = 2.f*(qx*qz + qw*qy);
    G[1][0] = 2.f*(qx*qy + qw*qz); G[1][1] = 1.f - 2.f*(qx*qx + qz*qz); G[1][2] = 2.f*(qy*qz - qw*qx);
    G[2][0] = 2.f*(qx*qz - qw*qy); G[2][1] = 2.f*(qy*qz + qw*qx); G[2][2] = 1.f - 2.f*(qx*qx + qy*qy);

    float s2[3] = {scale[n*3+0]*scale[n*3+0], scale[n*3+1]*scale[n*3+1], scale[n*3+2]*scale[n*3+2]};
    float C3[3][3];
    #pragma unroll
    for (int i = 0; i < 3; ++i)
        #pragma unroll
        for (int k = 0; k < 3; ++k)
            C3[i][k] = G[i][0]*G[k][0]*s2[0] + G[i][1]*G[k][1]*s2[1] + G[i][2]*G[k][2]*s2[2];

    // covc = Rcw * C3 * Rcw^T
    float Wm[3][3], Vc[3][3];
    #pragma unroll
    for (int i = 0; i < 3; ++i)
        #pragma unroll
        for (int j = 0; j < 3; ++j)
            Wm[i][j] = Rm[i][0]*C3[0][j] + Rm[i][1]*C3[1][j] + Rm[i][2]*C3[2][j];
    #pragma unroll
    for (int i = 0; i < 3; ++i)
        #pragma unroll
        for (int l = 0; l < 3; ++l)
            Vc[i][l] = Wm[i][0]*Rm[l][0] + Wm[i][1]*Rm[l][1] + Wm[i][2]*Rm[l][2];

    float rz = 1.0f / zc;
    float J0[3] = {fx*rz, 0.f, -fx*xc*rz*rz};
    float J1[3] = {0.f, fy*rz, -fy*yc*rz*rz};
    float u0[3], u1[3];
    #pragma unroll
    for (int k = 0; k < 3; ++k) {
        u0[k] = J0[0]*Vc[0][k] + J0[1]*Vc[1][k] + J0[2]*Vc[2][k];
        u1[k] = J1[0]*Vc[0][k] + J1[1]*Vc[1][k] + J1[2]*Vc[2][k];
    }
    float A2 = u0[0]*J0[0] + u0[1]*J0[1] + u0[2]*J0[2] + EPS2D;
    float B2 = u0[0]*J1[0] + u0[1]*J1[1] + u0[2]*J1[2];
    float C2 = u1[0]*J1[0] + u1[1]*J1[1] + u1[2]*J1[2] + EPS2D;
    float det = A2*C2 - B2*B2;
    float validf = ((zc > ZNEARF) && (zc < ZFARF) && (det > 1e-12f)) ? 1.f : 0.f;
    float inv_det = 1.0f / ((det > 1e-12f) ? det : 1.0f);
    float ca = C2 * inv_det, cb = -B2 * inv_det, cc = A2 * inv_det;
    float mx = fx * xc * rz + cx;
    float my = fy * yc * rz + cy;

    float* Ub = U + v * U_NPARAM * NGAUSS;
    Ub[0*NGAUSS+n] = mx;  Ub[1*NGAUSS+n] = my;
    Ub[2*NGAUSS+n] = ca;  Ub[3*NGAUSS+n] = cb;  Ub[4*NGAUSS+n] = cc;
    Ub[5*NGAUSS+n] = op;  Ub[6*NGAUSS+n] = validf;
    Ub[7*NGAUSS+n] = rgb[n*3+0]; Ub[8*NGAUSS+n] = rgb[n*3+1]; Ub[9*NGAUSS+n] = rgb[n*3+2];
    Ub[10*NGAUSS+n] = zc;
}

// ---------------------------------------------------------------------------
// Kernel 2: bitonic sort of (z, idx) per view, 2048 keys, one 1024-thread block
// ---------------------------------------------------------------------------
__global__ __launch_bounds__(1024) void sort_kernel(const float* __restrict__ U, int* __restrict__ sidx)
{
    __shared__ float key[NGAUSS];
    __shared__ int   val[NGAUSS];
    int v = blockIdx.x, tid = threadIdx.x;
    const float* z = U + v * U_NPARAM * NGAUSS + 10 * NGAUSS;
    for (int i = tid; i < NGAUSS; i += 1024) { key[i] = z[i]; val[i] = i; }
    for (unsigned k = 2; k <= NGAUSS; k <<= 1) {
        for (unsigned j = k >> 1; j >= 1; j >>= 1) {
            __syncthreads();
            unsigned l = ((tid & ~(j - 1u)) << 1) | (tid & (j - 1u));
            unsigned m = l | j;
            bool asc = ((l & k) == 0u);
            float kl = key[l], km = key[m];
            if (asc ? (kl > km) : (kl < km)) {
                key[l] = km; key[m] = kl;
                int tv = val[l]; val[l] = val[m]; val[m] = tv;
            }
        }
    }
    __syncthreads();
    for (int i = tid; i < NGAUSS; i += 1024) sidx[v * NGAUSS + i] = val[i];
}

// ---------------------------------------------------------------------------
// Kernel 3: gather into sorted order, precompute rank-6 bilinear coefficients.
// sigma(p,g) = v0*px^2 + v1*py^2 + v2*px*py + v3*px + v4*py + v5
// Coefficients pre-scaled by -log2(e) and opacity folded in, so the WMMA
// output is directly log2(alpha):  alpha = exp2(D).
// ---------------------------------------------------------------------------
__global__ void gather_kernel(const float* __restrict__ U, const int* __restrict__ sidx,
                              float* __restrict__ S)
{
    int gid = blockIdx.x * blockDim.x + threadIdx.x;
    if (gid >= NVIEW * NGAUSS) return;
    int v = gid / NGAUSS, i = gid % NGAUSS;
    int id = sidx[v * NGAUSS + i];
    const float* Ub = U + v * U_NPARAM * NGAUSS;
    float mx = Ub[0*NGAUSS+id], my = Ub[1*NGAUSS+id];
    float ca = Ub[2*NGAUSS+id], cb = Ub[3*NGAUSS+id], cc = Ub[4*NGAUSS+id];
    float op = Ub[5*NGAUSS+id], validf = Ub[6*NGAUSS+id];
    float v0 = 0.5f * ca, v1 = 0.5f * cc, v2 = cb;
    float v3 = -(ca * mx + cb * my);
    float v4 = -(cc * my + cb * mx);
    float v5 = 0.5f*ca*mx*mx + 0.5f*cc*my*my + cb*mx*my - __logf(fmaxf(op, 1e-30f));
    v0 *= NLOG2E; v1 *= NLOG2E; v2 *= NLOG2E;
    v3 *= NLOG2E; v4 *= NLOG2E; v5 *= NLOG2E;
    if (validf < 0.5f) v5 = -1e30f;   // invalid -> log2(alpha) = -inf -> alpha 0
    float* Sb = S + v * S_NPARAM * NGAUSS;
    Sb[0*NGAUSS+i] = v0; Sb[1*NGAUSS+i] = v1; Sb[2*NGAUSS+i] = v2;
    Sb[3*NGAUSS+i] = v3; Sb[4*NGAUSS+i] = v4; Sb[5*NGAUSS+i] = v5;
    Sb[6*NGAUSS+i] = Ub[7*NGAUSS+id];
    Sb[7*NGAUSS+i] = Ub[8*NGAUSS+id];
    Sb[8*NGAUSS+i] = Ub[9*NGAUSS+id];
    Sb[9*NGAUSS+i] = Ub[10*NGAUSS+id];
}

// ---------------------------------------------------------------------------
// Kernel 4: render. One block per (row, view); 8 waves x 16 pixels.
// Chunk staging: Tensor Data Mover (2D strided tile: 10 rows x 256 f32,
// row stride NGAUSS) lands contiguously in LDS == stage[p*256+i].
// Sigma tiles (16 px x 16 gaussians) via two chained V_WMMA_F32_16X16X4_F32.
// ---------------------------------------------------------------------------
__global__ __launch_bounds__(256) void render_kernel(const float* __restrict__ S,
                                                     float* __restrict__ out)
{
    __shared__ float stage[S_NPARAM * 256];   // sole LDS object -> LDS offset 0
    int tid  = threadIdx.x;
    int lane = tid & 31, wave = tid >> 5;
    int row  = blockIdx.x, view = blockIdx.y;
    int half = lane >> 4, p16 = lane & 15;

    float px = (float)(wave * 16 + p16) + 0.5f;
    float py = (float)row + 0.5f;

    // B[K][N=pixel]: VGPR0 holds K=0 (lanes 0-15) / K=2 (lanes 16-31); VGPR1 K=1/K=3.
    // WMMA0 features: [px^2, py^2, px*py, px]; WMMA1: [py, 1, 0, 0]
    v2f b0, b1;
    b0.x = half ? px * py : px * px;
    b0.y = half ? px      : py * py;
    b1.x = half ? 0.f : py;
    b1.y = half ? 0.f : 1.f;

    const float* Sb = S + view * S_NPARAM * NGAUSS;
    float Trun = 1.f, accR = 0.f, accG = 0.f, accB = 0.f, accW = 0.f, accD = 0.f;

    for (int chunk = 0; chunk < NGAUSS; chunk += 256) {
        __syncthreads();                       // previous chunk fully consumed
        if (wave == 0) {
            // Tensor DMA descriptor: tile 256 x 10 of f32, tensor row stride NGAUSS
            unsigned long long ga = (unsigned long long)(uintptr_t)(Sb + chunk);
            v4u g0;
            g0[0] = 1u;                                   // count=1, user mode, no gather
            g0[1] = 0u;                                   // lds_addr = offset of 'stage'
            g0[2] = (unsigned)(ga & 0xFFFFFFFFull);       // global_addr[31:0]
            g0[3] = (unsigned)(ga >> 32) | 0x80000000u;   // global_addr[56:32] | type=2
            v8i_t g1;
            g1[0] = (int)(2u << 16);                      // data_size = 4 bytes
            g1[1] = (int)(((unsigned)NGAUSS & 0xFFFFu) << 16);      // tensor_dim0[15:0]
            g1[2] = (int)(((unsigned)NGAUSS >> 16) | (10u << 16));  // dim0 hi | tensor_dim1 lo
            g1[3] = (int)(256u << 16);                    // tensor_dim1 hi | tile_dim0=256
            g1[4] = (int)10u;                             // tile_dim1=10, tile_dim2=0
            g1[5] = (int)NGAUSS;                          // tensor_dim0_stride[31:0]
            g1[6] = 0;                                    // stride hi | dim1_stride lo
            g1[7] = 0;
            v4i_t z4 = {0, 0, 0, 0};
#if __clang_major__ >= 23
            v8i_t z8 = {0, 0, 0, 0, 0, 0, 0, 0};
            __builtin_amdgcn_tensor_load_to_lds(g0, g1, z4, z4, z8, 0);
#else
            __builtin_amdgcn_tensor_load_to_lds(g0, g1, z4, z4, 0);
#endif
            __builtin_amdgcn_s_wait_tensorcnt(0);
        }
        __syncthreads();                       // data visible to all 8 waves

        for (int sub = 0; sub < 256; sub += 16) {
            int gl = sub + p16;   // this lane's gaussian (A-matrix row M = p16)
            // A[M=gaussian][K=feature], lane pairs hold K0,K1 / K2,K3
            v2f a0, a1;
            a0.x = half ? stage[2*256 + gl] : stage[0*256 + gl];
            a0.y = half ? stage[3*256 + gl] : stage[1*256 + gl];
            a1.x = half ? 0.f : stage[4*256 + gl];
            a1.y = half ? 0.f : stage[5*256 + gl];

            v8f c = {};
            c = __builtin_amdgcn_wmma_f32_16x16x4_f32(false, a1, false, b1, (short)0, c, false, false);
            c = __builtin_amdgcn_wmma_f32_16x16x4_f32(false, a0, false, b0, (short)0, c, false, false);

            // D layout: lanes<16 get gaussians sub..sub+7, lanes>=16 get sub+8..sub+15,
            // pixel = p16 in both halves. c[r] = log2(alpha).
            float av[8];
            float P = 1.f;
            #pragma unroll
            for (int r = 0; r < 8; ++r) {
                float al = __builtin_amdgcn_exp2f(c[r]);
                al = (al > A_MIN) ? fminf(al, A_MAX) : 0.f;
                av[r] = al;
                P *= (1.f - al);
            }
            float Pother = __shfl_xor(P, 16, 32);
            float Tst = half ? Trun * Pother : Trun;   // upper half starts after lower 8
            int base = sub + half * 8;
            #pragma unroll
            for (int r = 0; r < 8; ++r) {
                float w = Tst * av[r];
                Tst *= (1.f - av[r]);
                int m = base + r;
                accR += w * stage[6*256 + m];
                accG += w * stage[7*256 + m];
                accB += w * stage[8*256 + m];
                accW += w;
                accD += w * stage[9*256 + m];
            }
            Trun *= P * Pother;
        }
    }

    // combine lane-pair partial sums
    accR += __shfl_xor(accR, 16, 32);
    accG += __shfl_xor(accG, 16, 32);
    accB += __shfl_xor(accB, 16, 32);
    accW += __shfl_xor(accW, 16, 32);
    accD += __shfl_xor(accD, 16, 32);

    if (half == 0) {
        int x = wave * 16 + p16, y = row;
        int pix = (view * IMGH + y) * IMGW + x;
        float r3 = fminf(fmaxf(2.f * accR - 1.f, -1.f), 1.f);
        float g3 = fminf(fmaxf(2.f * accG - 1.f, -1.f), 1.f);
        float b3 = fminf(fmaxf(2.f * accB - 1.f, -1.f), 1.f);
        out[((view * 3 + 0) * IMGH + y) * IMGW + x] = r3;
        out[((view * 3 + 1) * IMGH + y) * IMGW + x] = g3;
        out[((view * 3 + 2) * IMGH + y) * IMGW + x] = b3;
        out[NVIEW * 3 * IMGH * IMGW + pix] = accW;
        out[NVIEW * 4 * IMGH * IMGW + pix] = accD / (accW + 1e-10f);
    }
}

// ---------------------------------------------------------------------------
extern "C" void kernel_launch(void* const* d_in, const int* in_sizes, int n_in,
                              void* d_out, int out_size, void* d_ws, size_t ws_size,
                              hipStream_t stream)
{
    (void)in_sizes; (void)n_in; (void)out_size; (void)ws_size;
    const float* xyz      = (const float*)d_in[0];
    const float* rgb      = (const float*)d_in[1];
    const float* scale    = (const float*)d_in[2];
    const float* rotation = (const float*)d_in[3];
    const float* opacity  = (const float*)d_in[4];
    const float* tin      = (const float*)d_in[5];
    const float* cov_t    = (const float*)d_in[6];
    const float* ms3      = (const float*)d_in[7];
    const float* w2cs     = (const float*)d_in[8];
    const float* Ks       = (const float*)d_in[9];
    const float* ts       = (const float*)d_in[10];

    float* ws   = (float*)d_ws;
    float* U    = ws + OFF_U;
    int*   SIDX = (int*)(ws + OFF_IDX);
    float* S    = ws + OFF_S;

    int tot = NVIEW * NGAUSS;
    prep_kernel<<<(tot + 255) / 256, 256, 0, stream>>>(xyz, rgb, scale, rotation, opacity,
                                                       tin, cov_t, ms3, w2cs, Ks, ts, U);
    sort_kernel<<<NVIEW, 1024, 0, stream>>>(U, SIDX);
    gather_kernel<<<(tot + 255) / 256, 256, 0, stream>>>(U, SIDX, S);
    render_kernel<<<dim3(IMGH, NVIEW), 256, 0, stream>>>(S, (float*)d_out);
}